// PrototypeReconstruction_1632087572493
// MI455X (gfx1250) — compile-verified
//
#include <hip/hip_runtime.h>

typedef __attribute__((ext_vector_type(2))) float v2f;
typedef __attribute__((ext_vector_type(8))) float v8f;

#define BDIM  64
#define HWSZ  4096                 // H*W
#define CDIM  128
#define PDIM  64
#define MROWS (BDIM * HWSZ)        // 262144 = B*H*W
#define NSLICE 16                  // split-K slices for z reduction

// ---------------------------------------------------------------------------
// Kernel 1: out[m][c] = sum_p s[m][p] * k[p][c]
// M=262144, K=64, N=128, via V_WMMA_F32_16X16X4_F32.
// One wave computes a 16x128 output stripe: 8 v8f accumulators,
// 16 K-steps of 4, A-fragment reused across all 8 N-tiles.
// ---------------------------------------------------------------------------
__global__ __launch_bounds__(256) void gemm_out_kernel(const float* __restrict__ s,
                                                       const float* __restrict__ kd,
                                                       float* __restrict__ out) {
    const int lane = threadIdx.x & 31;
    const int wave = threadIdx.x >> 5;
    const int half = lane >> 4;     // 0: lanes 0-15, 1: lanes 16-31
    const int l15  = lane & 15;
    const int m0   = (blockIdx.x * 8 + wave) * 16;

    // A fragment rows: lane l<16 -> row m0+l holding K=k0,k0+1 ; lanes 16-31 hold K=k0+2,k0+3
    const float* srow = s + (size_t)(m0 + l15) * PDIM + half * 2;

    v8f acc[8] = {};

    for (int kt = 0; kt < 16; ++kt) {
        const int k0 = kt * 4;
        v2f a;
        a.x = srow[k0];
        a.y = srow[k0 + 1];
        // B fragment: VGPR0 = row (k0 + half*2), VGPR1 = row (k0 + half*2 + 1), col = n0 + l15
        const float* kbase = kd + (size_t)(k0 + half * 2) * CDIM + l15;
#pragma unroll
        for (int j = 0; j < 8; ++j) {
            v2f b;
            b.x = kbase[j * 16];
            b.y = kbase[j * 16 + CDIM];
            acc[j] = __builtin_amdgcn_wmma_f32_16x16x4_f32(
                false, a, false, b, (short)0, acc[j], false, false);
        }
    }

    // C/D layout: VGPR r, lanes 0-15 -> row m0+r, lanes 16-31 -> row m0+r+8, col = n0 + l15
    float* obase = out + ((size_t)m0 + half * 8) * CDIM + l15;
#pragma unroll
    for (int j = 0; j < 8; ++j) {
#pragma unroll
        for (int r = 0; r < 8; ++r) {
            obase[(size_t)r * CDIM + j * 16] = acc[j][r];
        }
    }
}

// ---------------------------------------------------------------------------
// Kernel 2a: split-K reduction of z over hw.
// partial[b][slice][c]   = sum over 256 hw positions of z[b,hw,c]   (c = 0..127)
// partial[b][slice][128] = sum over those positions of z^2 (all c)
// Grid: 64*16 blocks x 256 threads. Fully deterministic (no atomics).
// ---------------------------------------------------------------------------
__global__ __launch_bounds__(256) void reduce_z_kernel(const float* __restrict__ z,
                                                       float* __restrict__ partial) {
    const int b     = blockIdx.x >> 4;
    const int slice = blockIdx.x & (NSLICE - 1);
    const int t     = threadIdx.x;
    const int c     = t & (CDIM - 1);
    const int sub   = t >> 7;   // 0 or 1: which of each hw-pair

    const float* zp = z + (size_t)b * HWSZ * CDIM + (size_t)(slice * 256 + sub) * CDIM + c;

    float acc = 0.f, acc2 = 0.f;
#pragma unroll 4
    for (int i = 0; i < 128; ++i) {
        float v = zp[(size_t)i * (2 * CDIM)];
        acc  += v;
        acc2 += v * v;
    }

    __shared__ float sC[256];
    __shared__ float sQ[256];
    sC[t] = acc;
    sQ[t] = acc2;
    __syncthreads();

    if (t < CDIM) {
        partial[((size_t)b * NSLICE + slice) * (CDIM + 1) + t] = sC[t] + sC[t + CDIM];
    }
    // tree-reduce z^2 over all 256 threads
    for (int stride = 128; stride > 0; stride >>= 1) {
        if (t < stride) sQ[t] += sQ[t + stride];
        __syncthreads();
    }
    if (t == 0) {
        partial[((size_t)b * NSLICE + slice) * (CDIM + 1) + CDIM] = sQ[0];
    }
}

// ---------------------------------------------------------------------------
// Kernel 2b: distance[b][p] = z2[b] - 2*cross[b][p] + 4096*k2[p]
// cross[b][p] = sum_c zc[b][c] * k[p][c]. Grid: 64 blocks x 64 threads.
// ---------------------------------------------------------------------------
__global__ __launch_bounds__(64) void distance_kernel(const float* __restrict__ partial,
                                                      const float* __restrict__ kd,
                                                      float* __restrict__ dist) {
    const int b = blockIdx.x;
    const int p = threadIdx.x;

    __shared__ float zc[CDIM];
    __shared__ float z2s;

    for (int c = p; c < CDIM; c += 64) {
        float v = 0.f;
#pragma unroll
        for (int sl = 0; sl < NSLICE; ++sl)
            v += partial[((size_t)b * NSLICE + sl) * (CDIM + 1) + c];
        zc[c] = v;
    }
    if (p == 0) {
        float v = 0.f;
#pragma unroll
        for (int sl = 0; sl < NSLICE; ++sl)
            v += partial[((size_t)b * NSLICE + sl) * (CDIM + 1) + CDIM];
        z2s = v;
    }
    __syncthreads();

    float cross = 0.f, k2 = 0.f;
    const float* kr = kd + (size_t)p * CDIM;
#pragma unroll 4
    for (int c = 0; c < CDIM; ++c) {
        float kv = kr[c];
        cross += zc[c] * kv;
        k2    += kv * kv;
    }
    dist[(size_t)b * PDIM + p] = z2s - 2.f * cross + (float)HWSZ * k2;
}

// ---------------------------------------------------------------------------
extern "C" void kernel_launch(void* const* d_in, const int* in_sizes, int n_in,
                              void* d_out, int out_size, void* d_ws, size_t ws_size,
                              hipStream_t stream) {
    const float* z  = (const float*)d_in[0];   // (B,H,W,C) f32
    const float* s  = (const float*)d_in[1];   // (B,H,W,P) f32
    const float* kd = (const float*)d_in[2];   // (P,1,1,C) f32 -> (P,C)

    float* out  = (float*)d_out;                          // (B,H,W,C)
    float* dist = out + (size_t)MROWS * CDIM;             // (B,P) appended
    float* partial = (float*)d_ws;                        // 64*16*129 floats = 516 KiB

    // GEMM: 262144 rows / (8 waves * 16 rows) = 2048 blocks
    gemm_out_kernel<<<MROWS / (8 * 16), 256, 0, stream>>>(s, kd, out);
    // z reduction: 64 b * 16 slices
    reduce_z_kernel<<<BDIM * NSLICE, 256, 0, stream>>>(z, partial);
    // finish distance
    distance_kernel<<<BDIM, 64, 0, stream>>>(partial, kd, dist);
}